// DynamicGraphLearner_48911087567222
// MI455X (gfx1250) — compile-verified
//
#include <hip/hip_runtime.h>
#include <math.h>

// Problem constants (match reference).
#define N_NODES 8192
#define HDIM    1024
#define KTOP    16
#define NSEC    16
#define SECLEN  (N_NODES / NSEC)   // 512 members per sector
#define NK      (N_NODES * KTOP)   // 131072 edges

typedef __attribute__((ext_vector_type(2))) float v2f;
typedef __attribute__((ext_vector_type(4))) float v4f;
typedef __attribute__((ext_vector_type(8))) float v8f;

// ---------------------------------------------------------------------------
// Stage 1: u = W^T a1, v = W^T a2   (reads W once: 4 MB)
// u[i] = sum_j W[j*H + i] * a[j] ; v[i] = sum_j W[j*H + i] * a[H + j]
// One thread per output column i; lane->i gives fully coalesced W reads.
// ---------------------------------------------------------------------------
__global__ __launch_bounds__(256) void dg_uv_kernel(const float* __restrict__ W,
                                                    const float* __restrict__ a,
                                                    float* __restrict__ u,
                                                    float* __restrict__ v) {
  const int i = blockIdx.x * blockDim.x + threadIdx.x;   // 0..HDIM-1
  const float* a2 = a + HDIM;
  float su = 0.f, sv = 0.f;
  for (int j = 0; j < HDIM; ++j) {
    const float w = W[(size_t)j * HDIM + i];
    su = fmaf(w, a[j],  su);   // a[j] is lane-uniform -> scalar load
    sv = fmaf(w, a2[j], sv);
  }
  u[i] = su;
  v[i] = sv;
}

// ---------------------------------------------------------------------------
// Stage 2: partial s_src = E @ u, s_dst = E @ v via V_WMMA_F32_16X16X4_F32.
//
// 1024 waves: wave = (tile, ksplit). Each wave owns 16 rows x 512 K of E.
// Per step of K=8: ONE global_load_b128 per lane covers the A-fragments of
// TWO WMMAs (half-wave 0 takes k..k+3, half-wave 1 takes k+4..k+7 -- the
// K-slot <-> global-k mapping inside a WMMA is arbitrary because we build the
// matching B fragment ourselves from u/v). A 3-deep rotating register
// pipeline keeps ~9 loads in flight so the s_wait before each WMMA covers
// only the oldest loads (latency overlapped), instead of loadcnt==0.
//
// A (16x4 f32): lane l -> row rowBase+(l&15); B (4x16): col 0 = u, col 1 = v,
// rest 0 (branchless cndmask selects keep EXEC all-ones as WMMA requires).
// Accumulator C: column N=0 is partial s_src, N=1 is partial s_dst.
//
// part layout (floats): [ksplit][{src,dst}][N_NODES]
//   psrc_k = part + ks*2*N ; pdst_k = part + ks*2*N + N
// ---------------------------------------------------------------------------
__global__ __launch_bounds__(128) void dg_score_kernel(const float* __restrict__ E,
                                                       const float* __restrict__ u,
                                                       const float* __restrict__ v,
                                                       float* __restrict__ part) {
  const int lane    = threadIdx.x & 31;
  const int wv      = blockIdx.x * (blockDim.x >> 5) + (threadIdx.x >> 5); // 0..1023
  const int tile    = wv >> 1;
  const int ks      = wv & 1;
  const int rowBase = tile * 16;
  const int m       = lane & 15;          // A: matrix row; B/C: column N
  const int half    = lane >> 4;          // 0 or 1
  const int h4      = half << 2;          // this half's K offset within a chunk
  const bool isU    = (m == 0);
  const bool isV    = (m == 1);
  const int kBeg    = ks * (HDIM / 2);    // 512-wide K range

  const float* eP = E + (size_t)(rowBase + m) * HDIM + kBeg + h4;
  const float* uP = u + kBeg + h4;
  const float* vP = v + kBeg + h4;

  // 3-stage pipeline prologue (B128 loads: 16B/lane, 512B/wave each).
  v4f aS[3], uS[3], vS[3];
  #pragma unroll
  for (int s = 0; s < 3; ++s) {
    aS[s] = *(const v4f*)(eP + 8 * s);
    uS[s] = *(const v4f*)(uP + 8 * s);
    vS[s] = *(const v4f*)(vP + 8 * s);
  }

  v8f c = {};
  const int NSTEPS = (HDIM / 2) / 8;      // 64 steps of K=8
  #pragma unroll 4
  for (int it = 0; it < NSTEPS; ++it) {
    const v4f aC = aS[0], uC = uS[0], vC = vS[0];
    aS[0] = aS[1]; uS[0] = uS[1]; vS[0] = vS[1];
    aS[1] = aS[2]; uS[1] = uS[2]; vS[1] = vS[2];
    const int kn = (it + 3 < NSTEPS) ? (it + 3) * 8 : 0;  // wrap: harmless refetch
    aS[2] = *(const v4f*)(eP + kn);
    uS[2] = *(const v4f*)(uP + kn);
    vS[2] = *(const v4f*)(vP + kn);

    // WMMA #0: K slots = {k, k+1} (half 0) / {k+4, k+5} (half 1)
    v2f a0; a0.x = aC.x; a0.y = aC.y;
    v2f b0;
    b0.x = isU ? uC.x : (isV ? vC.x : 0.f);
    b0.y = isU ? uC.y : (isV ? vC.y : 0.f);
    c = __builtin_amdgcn_wmma_f32_16x16x4_f32(false, a0, false, b0, (short)0, c,
                                              false, false);
    // WMMA #1: K slots = {k+2, k+3} (half 0) / {k+6, k+7} (half 1)
    v2f a1; a1.x = aC.z; a1.y = aC.w;
    v2f b1;
    b1.x = isU ? uC.z : (isV ? vC.z : 0.f);
    b1.y = isU ? uC.w : (isV ? vC.w : 0.f);
    c = __builtin_amdgcn_wmma_f32_16x16x4_f32(false, a1, false, b1, (short)0, c,
                                              false, false);
  }

  // C/D layout: VGPR r -> lanes 0-15: (M=r, N=lane); lanes16-31: (M=r+8, N=lane-16)
  float* psrc = part + ks * 2 * N_NODES;
  float* pdst = psrc + N_NODES;
  if (m == 0) {          // lanes 0 and 16: column N=0 -> partial s_src
    #pragma unroll
    for (int r = 0; r < 8; ++r) psrc[rowBase + half * 8 + r] = c[r];
  } else if (m == 1) {   // lanes 1 and 17: column N=1 -> partial s_dst
    #pragma unroll
    for (int r = 0; r < 8; ++r) pdst[rowBase + half * 8 + r] = c[r];
  }
}

// ---------------------------------------------------------------------------
// Stage 3: per-sector top-16 of s_dst (leaky_relu is monotone and s_src[i]
// is row-constant, so every row in a sector shares the same top-k indices).
// One block of 512 threads per sector; 16 rounds of LDS argmax reduction.
// Tie-break: smaller global index wins, matching jax.lax.top_k stability.
// Sums the two K-split partials on load.
// ---------------------------------------------------------------------------
__global__ __launch_bounds__(512) void dg_topk_kernel(const float* __restrict__ part,
                                                      float* __restrict__ topk_val,
                                                      int* __restrict__ topk_idx) {
  __shared__ float ov[SECLEN];
  __shared__ float rv[SECLEN];
  __shared__ int   ri[SECLEN];

  const int s = blockIdx.x;        // sector 0..15
  const int t = threadIdx.x;       // member slot 0..511
  const int j = s + NSEC * t;      // global node index in this sector

  ov[t] = part[N_NODES + j] + part[3 * N_NODES + j];   // s_dst[j]
  __syncthreads();

  for (int k = 0; k < KTOP; ++k) {
    rv[t] = ov[t];
    ri[t] = t;
    __syncthreads();
    for (int stride = SECLEN / 2; stride >= 1; stride >>= 1) {
      if (t < stride) {
        const float v2 = rv[t + stride];
        const int   i2 = ri[t + stride];
        if (v2 > rv[t] || (v2 == rv[t] && i2 < ri[t])) {
          rv[t] = v2;
          ri[t] = i2;
        }
      }
      __syncthreads();
    }
    const int winT = ri[0];
    if (t == 0) {
      topk_val[s * KTOP + k] = rv[0];
      topk_idx[s * KTOP + k] = s + NSEC * winT;
    }
    if (t == winT) ov[t] = -__builtin_inff();   // remove winner for next round
    __syncthreads();
  }
}

// ---------------------------------------------------------------------------
// Stage 4: emit edges. Row i, slot k:
//   src = topk_idx[sector(i)][k], dst = i,
//   weight = leaky_relu(s_src[i] + topk_dstval[sector(i)][k], 0.2) / 1.0
// d_out layout (flat, return order): edge_index[0][NK], edge_index[1][NK],
// edge_weight[NK]. Indices < 2^24 are exact in f32.
// ---------------------------------------------------------------------------
__global__ __launch_bounds__(256) void dg_emit_kernel(const float* __restrict__ part,
                                                      const float* __restrict__ topk_val,
                                                      const int* __restrict__ topk_idx,
                                                      float* __restrict__ out) {
  const int e = blockIdx.x * blockDim.x + threadIdx.x;   // 0..NK-1
  if (e >= NK) return;
  const int i = e >> 4;        // dst row
  const int k = e & 15;        // rank
  const int s = i & (NSEC - 1);

  const float ssrc = part[i] + part[2 * N_NODES + i];    // s_src[i]
  const float x = ssrc + topk_val[s * KTOP + k];
  const float w = (x >= 0.f) ? x : 0.2f * x;
  const int src = topk_idx[s * KTOP + k];

  out[e]          = (float)src;   // edge_index[0]
  out[NK + e]     = (float)i;     // edge_index[1]
  out[2 * NK + e] = w;            // edge_weight
}

// ---------------------------------------------------------------------------
// Launch. Workspace (floats):
//   u[1024] | v[1024] | part[4*8192] | topk_val[256] | topk_idx[256]  (~140 KB)
// ---------------------------------------------------------------------------
extern "C" void kernel_launch(void* const* d_in, const int* in_sizes, int n_in,
                              void* d_out, int out_size, void* d_ws, size_t ws_size,
                              hipStream_t stream) {
  const float* E = (const float*)d_in[0];   // embeddings [8192,1024] f32
  const float* W = (const float*)d_in[1];   // W          [1024,1024] f32
  const float* a = (const float*)d_in[2];   // a          [2048]      f32
  // d_in[3] = sector_mask: structure is analytic (i%16==j%16) -> unused.

  float* ws   = (float*)d_ws;
  float* u    = ws;
  float* v    = ws + HDIM;
  float* part = ws + 2 * HDIM;                        // 4*N_NODES floats
  float* tkv  = ws + 2 * HDIM + 4 * N_NODES;
  int*   tki  = (int*)(ws + 2 * HDIM + 4 * N_NODES + NSEC * KTOP);

  dg_uv_kernel<<<HDIM / 256, 256, 0, stream>>>(W, a, u, v);
  // 1024 waves (16-row tile x K-split 2), 4 waves/block -> 256 blocks.
  dg_score_kernel<<<256, 128, 0, stream>>>(E, u, v, part);
  dg_topk_kernel<<<NSEC, SECLEN, 0, stream>>>(part, tkv, tki);
  dg_emit_kernel<<<(NK + 255) / 256, 256, 0, stream>>>(part, tkv, tki, (float*)d_out);
}